// GNN_58033598104029
// MI455X (gfx1250) — compile-verified
//
#include <hip/hip_runtime.h>
#include <hip/hip_bf16.h>

typedef __attribute__((ext_vector_type(2))) float v2f;
typedef __attribute__((ext_vector_type(8))) float v8f;
typedef int v4i __attribute__((vector_size(16)));

#define HIDF 128
#define BN_EPS 1e-5f

#if __has_builtin(__builtin_amdgcn_global_load_async_to_lds_b128) && \
    __has_builtin(__builtin_amdgcn_s_wait_asynccnt)
#define HAVE_ASYNC_LDS 1
#else
#define HAVE_ASYNC_LDS 0
#endif

// ---------------- degree / normalization ----------------

__global__ __launch_bounds__(256) void k_deg_init(float* __restrict__ deg, int N) {
  int i = blockIdx.x * 256 + threadIdx.x;
  if (i < N) deg[i] = 1.0f;  // self loop
}

__global__ __launch_bounds__(256) void k_deg_edges(const int* __restrict__ dst,
                                                   float* __restrict__ deg, int E) {
  int e = blockIdx.x * 256 + threadIdx.x;
  if (e < E) atomicAdd(&deg[dst[e]], 1.0f);
}

__global__ __launch_bounds__(256) void k_dinv(const float* __restrict__ deg,
                                              float* __restrict__ dinv, int N) {
  int i = blockIdx.x * 256 + threadIdx.x;
  if (i < N) dinv[i] = rsqrtf(deg[i]);
}

// ---------------- fp32 WMMA GEMM with fused epilogues ----------------
// C_full[M x 128] = A[M x K] * W[K x 128]
// layer mode (agg != nullptr): Ts = acc*dinv[row]          -> C
//                              agg = Ts*dinv[row] + bias   -> agg (self-loop + bias)
// final mode (agg == nullptr): C = relu(acc + bias)
// block = 256 threads = 8 waves; block computes rows [16*bx, 16*bx+16), all 128 cols;
// wave w owns the 16-col tile at 16*w; K stepped by 4 via V_WMMA_F32_16X16X4_F32.

__global__ __launch_bounds__(256) void k_gemm_wmma(
    const float* __restrict__ A, int lda, const float* __restrict__ W, int K,
    float* __restrict__ C, const float* __restrict__ bias,
    const float* __restrict__ dinv, float* __restrict__ agg, int N) {
  __shared__ float As[16 * 384];  // 24 KB, supports K up to 384

  const int r0 = blockIdx.x * 16;
  const int tid = threadIdx.x;
  const int tile = 16 * K;  // K is a multiple of 4 -> tile multiple of 16B chunks

#if HAVE_ASYNC_LDS
  {
    // stage the 16 x K A-tile via the CDNA5 async copy engine (ASYNCcnt path),
    // one 16-byte chunk per lane; addresses are 16B-aligned (K % 4 == 0).
    float* Anc = const_cast<float*>(A);
    for (int ci = tid; (ci << 2) < tile; ci += 256) {
      int i = ci << 2;  // float index of chunk start
      int r = i / K;
      int k = i - r * K;
      int row = r0 + r;
      if (row >= N) row = N - 1;  // clamp (N%16==0 in practice)
      __builtin_amdgcn_global_load_async_to_lds_b128(
          (__attribute__((address_space(1))) v4i*)(Anc + (size_t)row * lda + k),
          (__attribute__((address_space(3))) v4i*)(&As[i]),
          /*offset=*/0, /*cpol=*/0);
    }
    __builtin_amdgcn_s_wait_asynccnt(0);
  }
#else
  for (int i = tid; i < tile; i += 256) {
    int r = i / K;
    int k = i - r * K;
    int row = r0 + r;
    if (row >= N) row = N - 1;
    As[i] = A[(size_t)row * lda + k];
  }
#endif
  __syncthreads();

  const int wave = tid >> 5;           // 0..7 -> column tile
  const int lane = tid & 31;
  const int col0 = wave << 4;
  const int m = lane & 15;             // A row / B col / store col within tile
  const int khalf = (lane >> 4) << 1;  // 0 for lanes 0-15, 2 for lanes 16-31

  v8f acc = {};
  for (int k = 0; k < K; k += 4) {
    v2f a, b;
    const float* ar = &As[m * K + k + khalf];
    a.x = ar[0];
    a.y = ar[1];
    const float* br = &W[(size_t)(k + khalf) * HIDF + col0 + m];
    b.x = br[0];
    b.y = br[HIDF];
    acc = __builtin_amdgcn_wmma_f32_16x16x4_f32(
        /*neg_a=*/false, a, /*neg_b=*/false, b,
        /*c_mod=*/(short)0, acc, /*reuse_a=*/false, /*reuse_b=*/false);
  }

  const int rowoff = (lane >> 4) << 3;  // lanes 16-31 hold M = r+8
  const int col = col0 + m;
  const float bc = bias[col];
  if (agg) {  // layer mode: emit scaled messages + self-loop-initialized agg
#pragma unroll
    for (int r = 0; r < 8; ++r) {
      int row = r0 + r + rowoff;
      if (row < N) {
        float di = dinv[row];
        float ts = acc[r] * di;
        size_t idx = (size_t)row * HIDF + col;
        C[idx] = ts;              // Ts[row][col] = T * dinv[row]
        agg[idx] = ts * di + bc;  // self loop: T * dinv^2 + bias
      }
    }
  } else {  // final mode: relu(acc + bias)
#pragma unroll
    for (int r = 0; r < 8; ++r) {
      int row = r0 + r + rowoff;
      if (row < N) C[(size_t)row * HIDF + col] = fmaxf(acc[r] + bc, 0.0f);
    }
  }
}

// ---------------- edge scatter: agg[d] += Ts[s] * dinv[d] ----------------

__global__ __launch_bounds__(256) void k_agg_edges(const float* __restrict__ Ts,
                                                   const int* __restrict__ src,
                                                   const int* __restrict__ dst,
                                                   const float* __restrict__ dinv,
                                                   float* __restrict__ agg, int E) {
  int e = (blockIdx.x * 256 + threadIdx.x) >> 5;  // one wave per edge
  if (e >= E) return;
  int lane = threadIdx.x & 31;
  int s = src[e], d = dst[e];
  float w = dinv[d];  // Ts already carries dinv[s]
  float4 t = *(const float4*)(Ts + (size_t)s * HIDF + lane * 4);
  float* a = agg + (size_t)d * HIDF + lane * 4;
  atomicAdd(a + 0, t.x * w);
  atomicAdd(a + 1, t.y * w);
  atomicAdd(a + 2, t.z * w);
  atomicAdd(a + 3, t.w * w);
}

// ---------------- ReLU + batchnorm ----------------

__global__ __launch_bounds__(256) void k_zero(float* __restrict__ p, int n) {
  int i = blockIdx.x * 256 + threadIdx.x;
  if (i < n) p[i] = 0.0f;
}

__global__ __launch_bounds__(128) void k_relu_colsum(float* __restrict__ agg,
                                                     float* __restrict__ sums,
                                                     float* __restrict__ sumsq, int N) {
  int c = threadIdx.x;  // 128 threads: one per channel
  float s = 0.0f, q = 0.0f;
  for (int n = blockIdx.x; n < N; n += gridDim.x) {
    size_t idx = (size_t)n * HIDF + c;
    float v = fmaxf(agg[idx], 0.0f);
    agg[idx] = v;  // keep relu'd value
    s += v;
    q += v * v;
  }
  atomicAdd(&sums[c], s);
  atomicAdd(&sumsq[c], q);
}

__global__ __launch_bounds__(128) void k_bn_stats(const float* __restrict__ sums,
                                                  const float* __restrict__ sumsq,
                                                  float* __restrict__ mean,
                                                  float* __restrict__ inv, int N) {
  int c = threadIdx.x;
  float invN = 1.0f / (float)N;
  float m = sums[c] * invN;
  float v = sumsq[c] * invN - m * m;
  mean[c] = m;
  inv[c] = rsqrtf(v + BN_EPS);
}

__global__ __launch_bounds__(256) void k_bn_apply(const float* __restrict__ r,
                                                  const float* __restrict__ g,
                                                  const float* __restrict__ be,
                                                  const float* __restrict__ mean,
                                                  const float* __restrict__ inv,
                                                  float* __restrict__ X, int ldx, int N) {
  int i = blockIdx.x * 256 + threadIdx.x;  // one float4 per thread
  if (i >= N * 32) return;
  int n = i >> 5;
  int c4 = (i & 31) << 2;
  float4 t = *(const float4*)(r + ((size_t)i << 2));
  float4 o;
  o.x = g[c4 + 0] * (t.x - mean[c4 + 0]) * inv[c4 + 0] + be[c4 + 0];
  o.y = g[c4 + 1] * (t.y - mean[c4 + 1]) * inv[c4 + 1] + be[c4 + 1];
  o.z = g[c4 + 2] * (t.z - mean[c4 + 2]) * inv[c4 + 2] + be[c4 + 2];
  o.w = g[c4 + 3] * (t.w - mean[c4 + 3]) * inv[c4 + 3] + be[c4 + 3];
  *(float4*)(X + (size_t)n * ldx + c4) = o;
}

// ---------------- host orchestration ----------------

extern "C" void kernel_launch(void* const* d_in, const int* in_sizes, int n_in,
                              void* d_out, int out_size, void* d_ws, size_t ws_size,
                              hipStream_t stream) {
  const float* x = (const float*)d_in[0];
  const int* ei = (const int*)d_in[1];
  // d_in[2] = batch (unused by reference computation)
  const float* W1 = (const float*)d_in[3];
  const float* b1 = (const float*)d_in[4];
  const float* g1 = (const float*)d_in[5];
  const float* be1 = (const float*)d_in[6];
  const float* W2 = (const float*)d_in[7];
  const float* b2 = (const float*)d_in[8];
  const float* g2 = (const float*)d_in[9];
  const float* be2 = (const float*)d_in[10];
  const float* W3 = (const float*)d_in[11];
  const float* b3 = (const float*)d_in[12];
  const float* g3 = (const float*)d_in[13];
  const float* be3 = (const float*)d_in[14];
  const float* Wl = (const float*)d_in[15];
  const float* bl = (const float*)d_in[16];

  const int N = in_sizes[0] / HIDF;
  const int E = in_sizes[1] / 2;
  const int* src = ei;
  const int* dst = ei + E;

  // workspace layout (floats)
  float* ws = (float*)d_ws;
  float* deg = ws;                    // N
  float* dinv = deg + N;              // N
  float* T = dinv + N;                // N*128 (scaled messages Ts)
  float* agg = T + (size_t)N * HIDF;  // N*128
  float* H = agg + (size_t)N * HIDF;  // N*384 (x1|x2|x3)
  float* sums = H + (size_t)N * 384;  // 128
  float* sumsq = sums + HIDF;         // 128
  float* mbuf = sumsq + HIDF;         // 128
  float* ibuf = mbuf + HIDF;          // 128

  const int bN = (N + 255) / 256;
  const int bE = (E + 255) / 256;
  const int bNE = (N * 32 + 255) / 256;  // one float4 per thread over N*128
  const int bEW = (E * 32 + 255) / 256;  // one wave per edge
  const int bG = (N + 15) / 16;          // GEMM row tiles

  // degrees + symmetric normalization
  k_deg_init<<<bN, 256, 0, stream>>>(deg, N);
  k_deg_edges<<<bE, 256, 0, stream>>>(dst, deg, E);
  k_dinv<<<bN, 256, 0, stream>>>(deg, dinv, N);

  // ---- layer 1: input x (lda=128) -> x1 = H+0 ----
  k_gemm_wmma<<<bG, 256, 0, stream>>>(x, HIDF, W1, HIDF, T, b1, dinv, agg, N);
  k_agg_edges<<<bEW, 256, 0, stream>>>(T, src, dst, dinv, agg, E);
  k_zero<<<1, 256, 0, stream>>>(sums, 256);  // sums + sumsq contiguous
  k_relu_colsum<<<512, 128, 0, stream>>>(agg, sums, sumsq, N);
  k_bn_stats<<<1, 128, 0, stream>>>(sums, sumsq, mbuf, ibuf, N);
  k_bn_apply<<<bNE, 256, 0, stream>>>(agg, g1, be1, mbuf, ibuf, H + 0, 384, N);

  // ---- layer 2: input x1 = H+0 (lda=384) -> x2 = H+128 ----
  k_gemm_wmma<<<bG, 256, 0, stream>>>(H + 0, 384, W2, HIDF, T, b2, dinv, agg, N);
  k_agg_edges<<<bEW, 256, 0, stream>>>(T, src, dst, dinv, agg, E);
  k_zero<<<1, 256, 0, stream>>>(sums, 256);
  k_relu_colsum<<<512, 128, 0, stream>>>(agg, sums, sumsq, N);
  k_bn_stats<<<1, 128, 0, stream>>>(sums, sumsq, mbuf, ibuf, N);
  k_bn_apply<<<bNE, 256, 0, stream>>>(agg, g2, be2, mbuf, ibuf, H + 128, 384, N);

  // ---- layer 3: input x2 = H+128 (lda=384) -> x3 = H+256 ----
  k_gemm_wmma<<<bG, 256, 0, stream>>>(H + 128, 384, W3, HIDF, T, b3, dinv, agg, N);
  k_agg_edges<<<bEW, 256, 0, stream>>>(T, src, dst, dinv, agg, E);
  k_zero<<<1, 256, 0, stream>>>(sums, 256);
  k_relu_colsum<<<512, 128, 0, stream>>>(agg, sums, sumsq, N);
  k_bn_stats<<<1, 128, 0, stream>>>(sums, sumsq, mbuf, ibuf, N);
  k_bn_apply<<<bNE, 256, 0, stream>>>(agg, g3, be3, mbuf, ibuf, H + 256, 384, N);

  // ---- final linear: out = relu(H[N x 384] @ Wl + bl) ----
  k_gemm_wmma<<<bG, 256, 0, stream>>>(H, 384, Wl, 384, (float*)d_out, bl,
                                      nullptr, nullptr, N);
}